// CategoryTheoryEngine_25907242729863
// MI455X (gfx1250) — compile-verified
//
#include <hip/hip_runtime.h>
#include <hip/hip_bf16.h>

typedef float v2f __attribute__((ext_vector_type(2)));
typedef float v8f __attribute__((ext_vector_type(8)));

#define N_CELLS 65536
#define IN_DIM  128
#define HID     256
#define OUT     128
#define MLPH    128
#define NMORPH  32
#define CPB     16          // cells per block in the main MLP kernel
#define W1LD    (IN_DIM + HID)   // 384, row stride of Wa1/Wg1

__device__ __forceinline__ v8f wmma4(v2f a, v2f b, v8f c) {
  // D = A(16x4 f32) * B(4x16 f32) + C(16x16 f32), full fp32 precision
  return __builtin_amdgcn_wmma_f32_16x16x4_f32(
      /*neg_a=*/false, a, /*neg_b=*/false, b,
      /*c_mod=*/(short)0, c, /*reuse_a=*/false, /*reuse_b=*/false);
}

// ---------------------------------------------------------------------------
// K0: project the broadcast x through the first-layer weights once.
// ax0[i] = ba1[i] + sum_k Wa1[i, k] * x[k]   (k < 128), same for g.
// ---------------------------------------------------------------------------
__global__ __launch_bounds__(128) void xproj_kernel(
    const float* __restrict__ x,
    const float* __restrict__ Wa1, const float* __restrict__ ba1,
    const float* __restrict__ Wg1, const float* __restrict__ bg1,
    float* __restrict__ ax0, float* __restrict__ gx0) {
  int i = threadIdx.x;
  float sa = ba1[i], sg = bg1[i];
  const float* wa = Wa1 + (size_t)i * W1LD;
  const float* wg = Wg1 + (size_t)i * W1LD;
  for (int k = 0; k < IN_DIM; ++k) {
    float xv = x[k];
    sa += wa[k] * xv;
    sg += wg[k] * xv;
  }
  ax0[i] = sa;
  gx0[i] = sg;
}

// ---------------------------------------------------------------------------
// K1 (dominant): per-cell 2-layer MLPs a and g via f32 WMMA.
//   H1a = relu(hiddens @ Wa1h^T + ax0)         [16 x 128]  (K = 256)
//   A   = H1a @ Wa2^T ; G likewise ; out = A - G + (ba2 - bg2)
//   tension[c] = mean(out[c,:]^2)
// Block: 128 threads = 4 waves; wave w owns N-tiles {32w, 32w+16} for both
// the a and g paths. All fragment loads are contiguous b64 per lane.
// ---------------------------------------------------------------------------
__global__ __launch_bounds__(128) void mlp_out_kernel(
    const float* __restrict__ hiddens,
    const float* __restrict__ Wa1, const float* __restrict__ Wa2,
    const float* __restrict__ ba2,
    const float* __restrict__ Wg1, const float* __restrict__ Wg2,
    const float* __restrict__ bg2,
    const float* __restrict__ ax0, const float* __restrict__ gx0,
    float* __restrict__ out_ws, float* __restrict__ tension_ws) {
  __shared__ float sh1a[CPB * MLPH];
  __shared__ float sh1g[CPB * MLPH];
  __shared__ float sout[CPB * OUT];
  __shared__ float tpart[CPB * 8];

  const int tid  = threadIdx.x;
  const int wave = tid >> 5;
  const int lane = tid & 31;
  const int lh   = lane >> 4;   // K-half selector (0: k0,k0+1 ; 1: k0+2,k0+3)
  const int l16  = lane & 15;
  const int cell0 = blockIdx.x * CPB;
  const int n0 = wave * 32;

  const v8f z8 = {0.f, 0.f, 0.f, 0.f, 0.f, 0.f, 0.f, 0.f};

  // ---------------- GEMM1: K = 256 over the hidden part of W1 ----------------
  v8f aA0 = z8, aA1 = z8, aG0 = z8, aG1 = z8;
  const float* Ap  = hiddens + (size_t)(cell0 + l16) * HID + 2 * lh;
  const float* Ba0 = Wa1 + (size_t)(n0      + l16) * W1LD + IN_DIM + 2 * lh;
  const float* Ba1 = Wa1 + (size_t)(n0 + 16 + l16) * W1LD + IN_DIM + 2 * lh;
  const float* Bg0 = Wg1 + (size_t)(n0      + l16) * W1LD + IN_DIM + 2 * lh;
  const float* Bg1 = Wg1 + (size_t)(n0 + 16 + l16) * W1LD + IN_DIM + 2 * lh;

#pragma unroll 4
  for (int k = 0; k < HID; k += 4) {
    v2f a  = *(const v2f*)(Ap  + k);
    v2f b0 = *(const v2f*)(Ba0 + k);
    v2f b1 = *(const v2f*)(Ba1 + k);
    v2f c0 = *(const v2f*)(Bg0 + k);
    v2f c1 = *(const v2f*)(Bg1 + k);
    aA0 = wmma4(a, b0, aA0);
    aA1 = wmma4(a, b1, aA1);
    aG0 = wmma4(a, c0, aG0);
    aG1 = wmma4(a, c1, aG1);
  }

  {   // bias (x-projection incl. b1) + ReLU -> LDS
    float ba0v = ax0[n0 + l16], ba1v = ax0[n0 + 16 + l16];
    float bg0v = gx0[n0 + l16], bg1v = gx0[n0 + 16 + l16];
#pragma unroll
    for (int r = 0; r < 8; ++r) {
      int row = r + 8 * lh;     // C/D layout: VGPR r, lanes 16-31 hold M = r+8
      sh1a[row * MLPH + n0 + l16]      = fmaxf(aA0[r] + ba0v, 0.f);
      sh1a[row * MLPH + n0 + 16 + l16] = fmaxf(aA1[r] + ba1v, 0.f);
      sh1g[row * MLPH + n0 + l16]      = fmaxf(aG0[r] + bg0v, 0.f);
      sh1g[row * MLPH + n0 + 16 + l16] = fmaxf(aG1[r] + bg1v, 0.f);
    }
  }
  __syncthreads();

  // ---------------- GEMM2: K = 128, A from LDS, B = W2 from global ----------
  v8f oA0 = z8, oA1 = z8, oG0 = z8, oG1 = z8;
  const float* W2a0 = Wa2 + (size_t)(n0      + l16) * MLPH + 2 * lh;
  const float* W2a1 = Wa2 + (size_t)(n0 + 16 + l16) * MLPH + 2 * lh;
  const float* W2g0 = Wg2 + (size_t)(n0      + l16) * MLPH + 2 * lh;
  const float* W2g1 = Wg2 + (size_t)(n0 + 16 + l16) * MLPH + 2 * lh;
  const float* A2a = &sh1a[l16 * MLPH + 2 * lh];
  const float* A2g = &sh1g[l16 * MLPH + 2 * lh];

#pragma unroll 4
  for (int k = 0; k < MLPH; k += 4) {
    v2f aa = *(const v2f*)(A2a + k);
    v2f ag = *(const v2f*)(A2g + k);
    v2f b0 = *(const v2f*)(W2a0 + k);
    v2f b1 = *(const v2f*)(W2a1 + k);
    v2f c0 = *(const v2f*)(W2g0 + k);
    v2f c1 = *(const v2f*)(W2g1 + k);
    oA0 = wmma4(aa, b0, oA0);
    oA1 = wmma4(aa, b1, oA1);
    oG0 = wmma4(ag, c0, oG0);
    oG1 = wmma4(ag, c1, oG1);
  }

  {   // out = A - G + (ba2 - bg2): to LDS (for tension) and to global ws
    float bo0 = ba2[n0 + l16] - bg2[n0 + l16];
    float bo1 = ba2[n0 + 16 + l16] - bg2[n0 + 16 + l16];
#pragma unroll
    for (int r = 0; r < 8; ++r) {
      int row = r + 8 * lh;
      float o0 = oA0[r] - oG0[r] + bo0;
      float o1 = oA1[r] - oG1[r] + bo1;
      sout[row * OUT + n0 + l16]      = o0;
      sout[row * OUT + n0 + 16 + l16] = o1;
      out_ws[(size_t)(cell0 + row) * OUT + n0 + l16]      = o0;
      out_ws[(size_t)(cell0 + row) * OUT + n0 + 16 + l16] = o1;
    }
  }
  __syncthreads();

  {   // tension[c] = mean(out^2): 8 threads per row, then combine
    int row = tid >> 3, seg = tid & 7;
    float s = 0.f;
#pragma unroll
    for (int q = 0; q < 16; ++q) {
      float v = sout[row * OUT + seg * 16 + q];
      s += v * v;
    }
    tpart[row * 8 + seg] = s;
  }
  __syncthreads();
  if (tid < CPB) {
    float s = 0.f;
#pragma unroll
    for (int q = 0; q < 8; ++q) s += tpart[tid * 8 + q];
    tension_ws[cell0 + tid] = s * (1.0f / OUT);
  }
}

// ---------------------------------------------------------------------------
// K2: per-chunk tension reduction (max + sum), 64 blocks x 1024 cells.
// ---------------------------------------------------------------------------
__global__ __launch_bounds__(256) void tred1_kernel(
    const float* __restrict__ t, float* __restrict__ pmax, float* __restrict__ psum) {
  __shared__ float sm[256], ss[256];
  int base = blockIdx.x * 1024;
  float m = -1e30f, s = 0.f;
  for (int c = threadIdx.x; c < 1024; c += 256) {
    float v = t[base + c];
    m = fmaxf(m, v);
    s += v;
  }
  sm[threadIdx.x] = m; ss[threadIdx.x] = s;
  __syncthreads();
  for (int st = 128; st > 0; st >>= 1) {
    if (threadIdx.x < st) {
      sm[threadIdx.x] = fmaxf(sm[threadIdx.x], sm[threadIdx.x + st]);
      ss[threadIdx.x] += ss[threadIdx.x + st];
    }
    __syncthreads();
  }
  if (threadIdx.x == 0) { pmax[blockIdx.x] = sm[0]; psum[blockIdx.x] = ss[0]; }
}

// K3: fold the 64 partials -> scal[0]=tmax, scal[1]=tsum
__global__ __launch_bounds__(64) void tred2_kernel(
    const float* __restrict__ pmax, const float* __restrict__ psum,
    float* __restrict__ scal) {
  __shared__ float sm[64], ss[64];
  int t = threadIdx.x;
  sm[t] = pmax[t]; ss[t] = psum[t];
  __syncthreads();
  for (int st = 32; st > 0; st >>= 1) {
    if (t < st) { sm[t] = fmaxf(sm[t], sm[t + st]); ss[t] += ss[t + st]; }
    __syncthreads();
  }
  if (t == 0) { scal[0] = sm[0]; scal[1] = ss[0]; }
}

// ---------------------------------------------------------------------------
// K4: partial softmax-weighted sums: pvec[b][j] = sum_c exp(t-tmax)*out[c][j]
// ---------------------------------------------------------------------------
__global__ __launch_bounds__(128) void wsum_kernel(
    const float* __restrict__ t, const float* __restrict__ out_ws,
    const float* __restrict__ scal, float* __restrict__ pvec, float* __restrict__ pz) {
  __shared__ float et[1024];
  __shared__ float zr[128];
  int b = blockIdx.x, j = threadIdx.x;
  int base = b * 1024;
  float tmax = scal[0];
  for (int c = j; c < 1024; c += 128) et[c] = expf(t[base + c] - tmax);
  __syncthreads();
  float zp = 0.f;
  for (int c = j; c < 1024; c += 128) zp += et[c];
  zr[j] = zp;
  __syncthreads();
  for (int st = 64; st > 0; st >>= 1) {
    if (j < st) zr[j] += zr[j + st];
    __syncthreads();
  }
  if (j == 0) pz[b] = zr[0];
  float acc = 0.f;
  for (int c = 0; c < 1024; ++c)
    acc += et[c] * out_ws[(size_t)(base + c) * OUT + j];
  pvec[b * OUT + j] = acc;
}

// ---------------------------------------------------------------------------
// K5: GRUCell + nat_w transform, only for the 64 gathered morph rows.
// slot 0..31 -> morph_src rows, slot 32..63 -> morph_tgt rows.
// ---------------------------------------------------------------------------
__global__ __launch_bounds__(256) void gru_kernel(
    const float* __restrict__ hiddens,
    const float* __restrict__ W_ih, const float* __restrict__ W_hh,
    const float* __restrict__ b_ih, const float* __restrict__ b_hh,
    const float* __restrict__ nat_w,
    const int* __restrict__ msrc, const int* __restrict__ mtgt,
    const int* __restrict__ step,
    const float* __restrict__ out_ws, const float* __restrict__ tension_ws,
    float* __restrict__ hid64) {
  __shared__ float mi[132];   // mem_in = [out row, tension]
  __shared__ float hr[HID];
  __shared__ float hn[HID];
  int slot = blockIdx.x;
  int idx = (slot < NMORPH) ? msrc[slot] : mtgt[slot - NMORPH];
  int t = threadIdx.x;
  if (t < OUT)          mi[t]   = out_ws[(size_t)idx * OUT + t];
  else if (t == OUT)    mi[OUT] = tension_ws[idx];
  hr[t] = hiddens[(size_t)idx * HID + t];
  __syncthreads();

  float gir = b_ih[t], giz = b_ih[HID + t], gin = b_ih[2 * HID + t];
  const float* wr = W_ih + (size_t)t            * (OUT + 1);
  const float* wz = W_ih + (size_t)(HID + t)    * (OUT + 1);
  const float* wn = W_ih + (size_t)(2 * HID + t)* (OUT + 1);
  for (int k = 0; k < OUT + 1; ++k) {
    float m = mi[k];
    gir += wr[k] * m; giz += wz[k] * m; gin += wn[k] * m;
  }
  float ghr = b_hh[t], ghz = b_hh[HID + t], ghn = b_hh[2 * HID + t];
  const float* vr = W_hh + (size_t)t             * HID;
  const float* vz = W_hh + (size_t)(HID + t)     * HID;
  const float* vn = W_hh + (size_t)(2 * HID + t) * HID;
  for (int k = 0; k < HID; ++k) {
    float h = hr[k];
    ghr += vr[k] * h; ghz += vz[k] * h; ghn += vn[k] * h;
  }
  float r  = 1.f / (1.f + expf(-(gir + ghr)));
  float z  = 1.f / (1.f + expf(-(giz + ghz)));
  float nn = tanhf(gin + r * ghn);
  float h  = (1.f - z) * nn + z * hr[t];
  hn[t] = h;
  __syncthreads();
  float res;
  if ((step[0] % 3) == 0) {    // hid = hid @ nat_w^T
    float s = 0.f;
    for (int k = 0; k < HID; ++k) s += nat_w[(size_t)t * HID + k] * hn[k];
    res = s;
  } else {
    res = h;
  }
  hid64[slot * HID + t] = res;
}

// ---------------------------------------------------------------------------
// K6: diff[i] = (limit - colimit)[i]
//   limit[i]   = sum_m sum_j mw[m,i,j] * hidS[m,j] / 32
//   colimit[i] = sum_m sum_j mw[m,j,i] * hidT[m,j] / 32
// ---------------------------------------------------------------------------
__global__ __launch_bounds__(256) void morph_kernel(
    const float* __restrict__ mw, const float* __restrict__ hid64,
    float* __restrict__ diffv) {
  __shared__ float red[256];
  int i = blockIdx.x, j = threadIdx.x;
  float la = 0.f, lb = 0.f;
  for (int m = 0; m < NMORPH; ++m) {
    la += mw[((size_t)m * HID + i) * HID + j] * hid64[m * HID + j];
    lb += mw[((size_t)m * HID + j) * HID + i] * hid64[(NMORPH + m) * HID + j];
  }
  red[j] = la - lb;
  __syncthreads();
  for (int s = 128; s > 0; s >>= 1) {
    if (j < s) red[j] += red[j + s];
    __syncthreads();
  }
  if (j == 0) diffv[i] = red[0] / (float)NMORPH;
}

// ---------------------------------------------------------------------------
// K7: finalize: d_out[0..127] = combined_out, d_out[128] = avg_tension.
// ---------------------------------------------------------------------------
__global__ __launch_bounds__(128) void final_kernel(
    const float* __restrict__ pvec, const float* __restrict__ pz,
    const float* __restrict__ diffv, const float* __restrict__ scal,
    float* __restrict__ out) {
  int j = threadIdx.x;
  float s = 0.f, z = 0.f;
  for (int b = 0; b < 64; ++b) { s += pvec[b * OUT + j]; z += pz[b]; }
  out[j] = s / z;
  if (j == 0) {
    float cat = 0.f;
    for (int i = 0; i < HID; ++i) { float d = diffv[i]; cat += d * d; }
    cat /= (float)HID;
    out[OUT] = scal[1] / (float)N_CELLS + 0.1f * cat;
  }
}

// ---------------------------------------------------------------------------
extern "C" void kernel_launch(void* const* d_in, const int* in_sizes, int n_in,
                              void* d_out, int out_size, void* d_ws, size_t ws_size,
                              hipStream_t stream) {
  (void)in_sizes; (void)n_in; (void)out_size; (void)ws_size;
  const float* x       = (const float*)d_in[0];
  const float* hiddens = (const float*)d_in[1];
  const float* Wa1     = (const float*)d_in[2];
  const float* ba1     = (const float*)d_in[3];
  const float* Wa2     = (const float*)d_in[4];
  const float* ba2     = (const float*)d_in[5];
  const float* Wg1     = (const float*)d_in[6];
  const float* bg1     = (const float*)d_in[7];
  const float* Wg2     = (const float*)d_in[8];
  const float* bg2     = (const float*)d_in[9];
  const float* W_ih    = (const float*)d_in[10];
  const float* W_hh    = (const float*)d_in[11];
  const float* b_ih    = (const float*)d_in[12];
  const float* b_hh    = (const float*)d_in[13];
  const float* nat_w   = (const float*)d_in[14];
  const float* morph_w = (const float*)d_in[15];
  const int*   msrc    = (const int*)d_in[16];
  const int*   mtgt    = (const int*)d_in[17];
  const int*   step    = (const int*)d_in[18];
  float* out = (float*)d_out;

  // workspace layout (floats)
  float* ws = (float*)d_ws;
  size_t o = 0;
  float* out_ws  = ws + o; o += (size_t)N_CELLS * OUT;   // 8,388,608
  float* tension = ws + o; o += N_CELLS;
  float* ax0     = ws + o; o += MLPH;
  float* gx0     = ws + o; o += MLPH;
  float* hid64   = ws + o; o += 2 * NMORPH * HID;        // 16,384
  float* diffv   = ws + o; o += HID;
  float* pmax    = ws + o; o += 64;
  float* psum    = ws + o; o += 64;
  float* pz      = ws + o; o += 64;
  float* pvec    = ws + o; o += 64 * OUT;
  float* scal    = ws + o; o += 2;

  xproj_kernel<<<1, 128, 0, stream>>>(x, Wa1, ba1, Wg1, bg1, ax0, gx0);
  mlp_out_kernel<<<N_CELLS / CPB, 128, 0, stream>>>(
      hiddens, Wa1, Wa2, ba2, Wg1, Wg2, bg2, ax0, gx0, out_ws, tension);
  tred1_kernel<<<64, 256, 0, stream>>>(tension, pmax, psum);
  tred2_kernel<<<1, 64, 0, stream>>>(pmax, psum, scal);
  wsum_kernel<<<64, 128, 0, stream>>>(tension, out_ws, scal, pvec, pz);
  gru_kernel<<<2 * NMORPH, 256, 0, stream>>>(
      hiddens, W_ih, W_hh, b_ih, b_hh, nat_w, msrc, mtgt, step,
      out_ws, tension, hid64);
  morph_kernel<<<HID, 256, 0, stream>>>(morph_w, hid64, diffv);
  final_kernel<<<1, 128, 0, stream>>>(pvec, pz, diffv, scal, out);
}